// GIN_6030134083939
// MI455X (gfx1250) — compile-verified
//
#include <hip/hip_runtime.h>

typedef float v2f __attribute__((ext_vector_type(2)));
typedef float v8f __attribute__((ext_vector_type(8)));

#define HID 128
#define NNODES 50000
#define NEDGES 800000
#define NGRAPHS 64

// Transposed W tile in LDS: wt[n][k], row stride 132 dwords (132 % 64 == 4)
// -> the wave's 16 B-operand ds_load_b64's hit 16 disjoint bank pairs, and
// (k, k+1) is contiguous so each B operand is a single aligned b64 load.
#define WSTRIDE 132

static __device__ __forceinline__ void atomic_add_f32(float* p, float v) {
  (void)__hip_atomic_fetch_add(p, v, __ATOMIC_RELAXED,
                               __HIP_MEMORY_SCOPE_AGENT);
}

// ---------------------------------------------------------------------------
// Fused GEMM: Y = X @ W + b, optionally followed by LayerNorm (over the 128
// feature dim) + ReLU. One wave computes a 16x128 output tile using
// V_WMMA_F32_16X16X4_F32 (exact f32 math, matches the f32 reference).
//
// W (128x128, 64KB) is staged TRANSPOSED into LDS once per 256-thread block
// via the CDNA5 async global->LDS path (GLOBAL_LOAD_ASYNC_TO_LDS_B32 with a
// per-lane transposing LDS destination, tracked by ASYNCcnt); all 8 waves'
// B operands are then single ds_load_b64's from LDS.
//
// A operand (16x4 f32): lanes 0-15 hold rows M=0..15 with K={0,1} in the two
// VGPRs; lanes 16-31 hold the same rows with K={2,3}.
// B operand (4x16 f32): mirrored with N in place of M.
// C/D (16x16 f32): VGPR v, lane L -> row = v + 8*(L>>4), col = L&15.
// ---------------------------------------------------------------------------
template <bool LNRELU>
__global__ __launch_bounds__(256) void gemm128_kernel(
    const float* __restrict__ X,     // [M, 128]
    const float* __restrict__ W,     // [128, 128] row-major (in, out)
    const float* __restrict__ bias,  // [128]
    const float* __restrict__ gam,   // [128] (LN gain)    -- only if LNRELU
    const float* __restrict__ bet,   // [128] (LN bias)    -- only if LNRELU
    float* __restrict__ Y,           // [M, 128]
    int M) {
  __shared__ float wsh[HID * WSTRIDE];  // 67,584 B, holds W transposed

  const int lane = threadIdx.x & 31;
  const int wave = threadIdx.x >> 5;
  const int tile = blockIdx.x * 8 + wave;  // 16-row tile index
  const int half = lane >> 4;              // 0 or 1
  const int l16 = lane & 15;

  // ---- async-stage W transposed into LDS ----
  // Thread t owns LDS row n = t & 127 and K-phase k0 = t >> 7; it streams
  // elements W[k][n] for k = k0, k0+2, ..., k0+126 (64 b32 async copies).
  // Global side: a wave's 32 lanes read 32 consecutive floats of one W row
  // (fully coalesced); LDS side: transposing scatter, stride WSTRIDE rows.
  {
    const unsigned lds0 = (unsigned)(size_t)(&wsh[0]);
    const int n = threadIdx.x & 127;
    const int k0 = threadIdx.x >> 7;  // 0 or 1
    unsigned lds = lds0 + (unsigned)((n * WSTRIDE + k0) * 4);
    const float* gp = W + (size_t)k0 * HID + n;  // &W[k0][n]
#pragma unroll
    for (int j = 0; j < 64; ++j) {
      asm volatile("global_load_async_to_lds_b32 %0, %1, off"
                   :
                   : "v"(lds), "v"(gp)
                   : "memory");
      lds += 8;        // k += 2 in transposed row
      gp += 2 * HID;   // two W rows down
    }
    asm volatile("s_wait_asynccnt 0x0" ::: "memory");
  }
  __syncthreads();

  if (tile * 16 < M) {  // wave-uniform: EXEC stays all-1s inside
    const int m0 = tile * 16;

    const v8f vzero = {0.f, 0.f, 0.f, 0.f, 0.f, 0.f, 0.f, 0.f};
    v8f acc[8];
#pragma unroll
    for (int nt = 0; nt < 8; ++nt) acc[nt] = vzero;

    const float* arow = X + (size_t)(m0 + l16) * HID;
#pragma unroll 4
    for (int kk = 0; kk < 32; ++kk) {
      const int k = kk * 4 + half * 2;
      const v2f a = *(const v2f*)(arow + k);          // b64, 8B aligned
      const float* wb = &wsh[l16 * WSTRIDE + k];      // wt[l16][k]
#pragma unroll
      for (int nt = 0; nt < 8; ++nt) {
        // wt[nt*16 + l16][k .. k+1] -> one aligned ds_load_b64
        const v2f b = *(const v2f*)(wb + nt * 16 * WSTRIDE);
        acc[nt] = __builtin_amdgcn_wmma_f32_16x16x4_f32(
            false, a, false, b, (short)0, acc[nt], false, false);
      }
    }

    // bias add (per output column)
#pragma unroll
    for (int nt = 0; nt < 8; ++nt) {
      const float bv = bias[nt * 16 + l16];
#pragma unroll
      for (int v = 0; v < 8; ++v) acc[nt][v] += bv;
    }

    if (LNRELU) {
      // Each 16-lane half-wave holds full rows: row = v + 8*half,
      // columns = nt*16 + l16 across the 8 accumulators.
#pragma unroll
      for (int v = 0; v < 8; ++v) {
        float s = 0.f, q = 0.f;
#pragma unroll
        for (int nt = 0; nt < 8; ++nt) {
          const float t = acc[nt][v];
          s += t;
          q += t * t;
        }
        // Reduce across the 16 lanes of this half-wave (xor masks < 16 never
        // cross the half boundary on wave32).
#pragma unroll
        for (int m = 1; m < 16; m <<= 1) {
          s += __shfl_xor(s, m, 32);
          q += __shfl_xor(q, m, 32);
        }
        const float mean = s * (1.f / HID);
        const float var = q * (1.f / HID) - mean * mean;
        const float inv = rsqrtf(var + 1e-5f);
#pragma unroll
        for (int nt = 0; nt < 8; ++nt) {
          float t = (acc[nt][v] - mean) * inv;
          t = t * gam[nt * 16 + l16] + bet[nt * 16 + l16];
          acc[nt][v] = t > 0.f ? t : 0.f;
        }
      }
    }

    // store tile
#pragma unroll
    for (int nt = 0; nt < 8; ++nt) {
#pragma unroll
      for (int v = 0; v < 8; ++v) {
        Y[(size_t)(m0 + v + 8 * half) * HID + nt * 16 + l16] = acc[nt][v];
      }
    }
  }
}

// h = x (seed for GIN sum: h_i = x_i + sum_j x_j)
__global__ __launch_bounds__(256) void copy_f4_kernel(
    const float4* __restrict__ in, float4* __restrict__ out, int n4) {
  const int i = blockIdx.x * blockDim.x + threadIdx.x;
  if (i < n4) out[i] = in[i];
}

// One wave per edge: coalesced 512B gather of x[src], atomic scatter to h[dst]
__global__ __launch_bounds__(256) void edge_scatter_kernel(
    const float* __restrict__ x, const int* __restrict__ src,
    const int* __restrict__ dst, float* __restrict__ h, int nE) {
  const int t = blockIdx.x * blockDim.x + threadIdx.x;
  const int e = t >> 5;
  const int lane = t & 31;
  if (e >= nE) return;
  const int s = src[e];
  const int d = dst[e];
  const float4 v = *(const float4*)(x + (size_t)s * HID + lane * 4);
  float* p = h + (size_t)d * HID + lane * 4;
  atomic_add_f32(p + 0, v.x);
  atomic_add_f32(p + 1, v.y);
  atomic_add_f32(p + 2, v.z);
  atomic_add_f32(p + 3, v.w);
}

__global__ __launch_bounds__(256) void zero_kernel(float* __restrict__ p,
                                                   int n) {
  const int i = blockIdx.x * blockDim.x + threadIdx.x;
  if (i < n) p[i] = 0.f;
}

// One wave per node: sums[batch[i]] += x[i]; cnt[batch[i]] += 1
__global__ __launch_bounds__(256) void pool_accum_kernel(
    const float* __restrict__ x, const int* __restrict__ batch,
    float* __restrict__ sums, float* __restrict__ cnt, int N) {
  const int t = blockIdx.x * blockDim.x + threadIdx.x;
  const int node = t >> 5;
  const int lane = t & 31;
  if (node >= N) return;
  const int g = batch[node];
  const float4 v = *(const float4*)(x + (size_t)node * HID + lane * 4);
  float* p = sums + (size_t)g * HID + lane * 4;
  atomic_add_f32(p + 0, v.x);
  atomic_add_f32(p + 1, v.y);
  atomic_add_f32(p + 2, v.z);
  atomic_add_f32(p + 3, v.w);
  if (lane == 0) atomic_add_f32(&cnt[g], 1.f);
}

__global__ __launch_bounds__(256) void pool_div_kernel(
    const float* __restrict__ sums, const float* __restrict__ cnt,
    float* __restrict__ out) {
  const int i = blockIdx.x * blockDim.x + threadIdx.x;
  if (i >= NGRAPHS * HID) return;
  const float c = cnt[i / HID];
  out[i] = sums[i] / (c > 1.f ? c : 1.f);
}

extern "C" void kernel_launch(void* const* d_in, const int* in_sizes, int n_in,
                              void* d_out, int out_size, void* d_ws,
                              size_t ws_size, hipStream_t stream) {
  // ---- inputs in setup_inputs() dict (insertion) order, pytree-flattened ---
  const float* x = (const float*)d_in[0];
  const int* edge = (const int*)d_in[1];  // [2, NEDGES]: row0=src, row1=dst
  const int* src = edge;
  const int* dst = edge + NEDGES;
  const int* batch = (const int*)d_in[2];
  // enc: lin[(W0,b0),(W1,b1)], ln[(g0,be0)]
  const float* encW0 = (const float*)d_in[3];
  const float* encB0 = (const float*)d_in[4];
  const float* encW1 = (const float*)d_in[5];
  const float* encB1 = (const float*)d_in[6];
  const float* encG0 = (const float*)d_in[7];
  const float* encBe0 = (const float*)d_in[8];
  // convs: 4 x [W0,b0,W1,b1,g0,be0,g1,be1] starting at index 9
  // dec: lin[(W0,b0),(W1,b1)], ln[(g0,be0)] starting at index 41
  const float* decW0 = (const float*)d_in[41];
  const float* decB0 = (const float*)d_in[42];
  const float* decW1 = (const float*)d_in[43];
  const float* decB1 = (const float*)d_in[44];
  const float* decG0 = (const float*)d_in[45];
  const float* decBe0 = (const float*)d_in[46];

  // ---- workspace layout ----
  const size_t NH = (size_t)NNODES * HID;
  float* buf0 = (float*)d_ws;         // activations ping
  float* buf1 = buf0 + NH;            // activations pong
  float* buf2 = buf1 + NH;            // h = x + agg / pooled
  float* sums = buf2 + NH;            // [NGRAPHS, HID]
  float* cnt = sums + NGRAPHS * HID;  // [NGRAPHS], contiguous after sums

  const dim3 blk(256);
  const int gemmBlocks = ((NNODES / 16) + 7) / 8;      // 391
  const int copyBlocks = (int)((NH / 4 + 255) / 256);  // 6250
  const int edgeBlocks = (NEDGES * 32 + 255) / 256;    // 100000
  const int poolBlocks = (NNODES * 32 + 255) / 256;    // 6250

  // encoder: LN+ReLU after layer 0, plain linear for layer 1
  gemm128_kernel<true><<<gemmBlocks, blk, 0, stream>>>(
      x, encW0, encB0, encG0, encBe0, buf0, NNODES);
  gemm128_kernel<false><<<gemmBlocks, blk, 0, stream>>>(
      buf0, encW1, encB1, nullptr, nullptr, buf1, NNODES);

  // GIN conv layers (eps=0): x <- MLP(x + segment_sum(x[src] -> dst))
  for (int l = 0; l < 4; ++l) {
    const float* W0 = (const float*)d_in[9 + 8 * l + 0];
    const float* B0 = (const float*)d_in[9 + 8 * l + 1];
    const float* W1 = (const float*)d_in[9 + 8 * l + 2];
    const float* B1 = (const float*)d_in[9 + 8 * l + 3];
    const float* G0 = (const float*)d_in[9 + 8 * l + 4];
    const float* Be0 = (const float*)d_in[9 + 8 * l + 5];
    const float* G1 = (const float*)d_in[9 + 8 * l + 6];
    const float* Be1 = (const float*)d_in[9 + 8 * l + 7];

    copy_f4_kernel<<<copyBlocks, blk, 0, stream>>>(
        (const float4*)buf1, (float4*)buf2, (int)(NH / 4));
    edge_scatter_kernel<<<edgeBlocks, blk, 0, stream>>>(buf1, src, dst, buf2,
                                                        NEDGES);
    gemm128_kernel<true><<<gemmBlocks, blk, 0, stream>>>(
        buf2, W0, B0, G0, Be0, buf0, NNODES);
    gemm128_kernel<true><<<gemmBlocks, blk, 0, stream>>>(
        buf0, W1, B1, G1, Be1, buf1, NNODES);
  }

  // global mean pool per graph
  zero_kernel<<<(NGRAPHS * HID + NGRAPHS + 255) / 256, blk, 0, stream>>>(
      sums, NGRAPHS * HID + NGRAPHS);
  pool_accum_kernel<<<poolBlocks, blk, 0, stream>>>(buf1, batch, sums, cnt,
                                                    NNODES);
  pool_div_kernel<<<(NGRAPHS * HID + 255) / 256, blk, 0, stream>>>(sums, cnt,
                                                                   buf2);

  // decoder on [NGRAPHS, HID]
  gemm128_kernel<true><<<1, blk, 0, stream>>>(buf2, decW0, decB0, decG0,
                                              decBe0, buf0, NGRAPHS);
  gemm128_kernel<false><<<1, blk, 0, stream>>>(buf0, decW1, decB1, nullptr,
                                               nullptr, (float*)d_out,
                                               NGRAPHS);
}